// ROIHead_60662118089266
// MI455X (gfx1250) — compile-verified
//
#include <hip/hip_runtime.h>
#include <cstdint>

// ROI head: matcher (IoU max/argmax vs 64 gt) + BoxCoder encode (targets) +
// BoxCoder decode (deltas). One thread per proposal; gt boxes async-staged
// into LDS via the CDNA5 async-copy path (global_load_async_to_lds_b128 +
// s_wait_asynccnt), then broadcast-read in the 64-iteration IoU loop.

#define MAX_GT 64

__global__ __launch_bounds__(256) void roihead_kernel(
    const float* __restrict__ proposals,  // [B,N,4]
    const float* __restrict__ gt_boxes,   // [B,M,4]
    const float* __restrict__ deltas,     // [B,N,4]
    float* __restrict__ out_decoded,      // [B,N,4]
    float* __restrict__ out_targets,      // [B,N,4]
    float* __restrict__ out_matches,      // [B,N]  (int values stored as float)
    int N, int M)
{
    __shared__ float s_gt[MAX_GT * 4];

    const int b   = blockIdx.y;
    const int tid = threadIdx.x;

    // ---- CDNA5 async stage: this image's gt boxes (M * 16B) -> LDS ----
    if (tid < M) {
        unsigned long long gaddr =
            (unsigned long long)(uintptr_t)(gt_boxes + ((size_t)b * M + tid) * 4);
        // Low 32 bits of the flat address of a __shared__ object are the LDS
        // byte address.
        unsigned lds_addr = (unsigned)(uintptr_t)(&s_gt[tid * 4]);
        asm volatile("global_load_async_to_lds_b128 %0, %1, off"
                     :
                     : "v"(lds_addr), "v"(gaddr)
                     : "memory");
    }
    // Issuing waves drain their ASYNCcnt before signaling the barrier;
    // non-issuing waves pass trivially.
    asm volatile("s_wait_asynccnt 0" ::: "memory");
    __syncthreads();   // s_barrier_signal/-wait -1: LDS now visible to all waves

    const int i = blockIdx.x * blockDim.x + tid;
    if (i >= N) return;

    const size_t gi = (size_t)b * N + i;

    // ---- proposal anchor terms ----
    const float4 p  = reinterpret_cast<const float4*>(proposals)[gi];
    const float aw  = p.z - p.x;
    const float ah  = p.w - p.y;
    const float ax  = p.x + 0.5f * aw;
    const float ay  = p.y + 0.5f * ah;
    const float areap = aw * ah;

    // ---- matcher: max/argmax IoU over gt axis (first occurrence of max) ----
    float best = -1.0f;
    int   bidx = 0;
    const float4* sg = reinterpret_cast<const float4*>(s_gt);
    #pragma unroll 4
    for (int m = 0; m < M; ++m) {
        const float4 g = sg[m];                 // LDS broadcast (all lanes same addr)
        const float areag = (g.z - g.x) * (g.w - g.y);
        float iw = fminf(p.z, g.z) - fmaxf(p.x, g.x);
        float ih = fminf(p.w, g.w) - fmaxf(p.y, g.y);
        iw = fmaxf(iw, 0.0f);
        ih = fmaxf(ih, 0.0f);
        const float inter = iw * ih;
        const float iou = inter / (areag + areap - inter);
        if (iou > best) { best = iou; bidx = m; }
    }
    // FG_THRESH == BG_THRESH == 0.5 -> BETWEEN(-2) branch is unreachable.
    const int match = (best >= 0.5f) ? bidx : -1;

    // ---- encode targets against matched (index-clamped) gt ----
    const float4 mg = sg[match < 0 ? 0 : match];
    const float gw = fmaxf(mg.z - mg.x, 1.0f);
    const float gh = fmaxf(mg.w - mg.y, 1.0f);
    const float gx = mg.x + 0.5f * gw;
    const float gy = mg.y + 0.5f * gh;
    float4 t;
    t.x = ((gx - ax) / aw) / 0.1f;              // divide by weights, like reference
    t.y = ((gy - ay) / ah) / 0.1f;
    t.z = logf(gw / aw) / 0.2f;
    t.w = logf(gh / ah) / 0.2f;

    // ---- decode regression deltas ----
    const float4 d = reinterpret_cast<const float4*>(deltas)[gi];
    const float sx = d.x * 0.1f;
    const float sy = d.y * 0.1f;
    const float sw = d.z * 0.2f;
    const float sh = d.w * 0.2f;
    const float cx = ax + sx * aw;
    const float cy = ay + sy * ah;
    const float ww = expf(sw) * aw;
    const float hh = expf(sh) * ah;
    float4 dec;
    dec.x = cx - 0.5f * ww;
    dec.y = cy - 0.5f * hh;
    dec.z = cx + 0.5f * ww;
    dec.w = cy + 0.5f * hh;

    // ---- coalesced B128 stores ----
    reinterpret_cast<float4*>(out_decoded)[gi] = dec;
    reinterpret_cast<float4*>(out_targets)[gi] = t;
    out_matches[gi] = (float)match;              // -1 or 0..63: exact in f32
}

extern "C" void kernel_launch(void* const* d_in, const int* in_sizes, int n_in,
                              void* d_out, int out_size, void* d_ws, size_t ws_size,
                              hipStream_t stream) {
    const float* proposals = (const float*)d_in[0];   // [B,N,4]
    const float* gt_boxes  = (const float*)d_in[1];   // [B,M,4]
    const float* deltas    = (const float*)d_in[2];   // [B,N,4]

    const int B = 4;                                  // fixed by the reference setup
    const int N = in_sizes[0] / (B * 4);
    const int M = in_sizes[1] / (B * 4);              // 64 (fits MAX_GT)

    float* out       = (float*)d_out;
    float* out_dec   = out;                                   // [B,N,4]
    float* out_tgt   = out + (size_t)B * N * 4;               // [B,N,4]
    float* out_match = out + (size_t)B * N * 8;               // [B,N]

    dim3 block(256);
    dim3 grid((N + 255) / 256, B);
    hipLaunchKernelGGL(roihead_kernel, grid, block, 0, stream,
                       proposals, gt_boxes, deltas,
                       out_dec, out_tgt, out_match, N, M);
}